// GuidedFilter_9208409882768
// MI455X (gfx1250) — compile-verified
//
#include <hip/hip_runtime.h>
#include <cstdint>
#include <cstddef>

#define DEV __device__ __forceinline__

namespace gf {

constexpr int Bn = 4;
constexpr int Hn = 1024;
constexpr int Wn = 1024;
constexpr int Cn = 4;
constexpr int Rr = 40;                 // box radius (reference r=40)
constexpr int WC = Wn * Cn;            // floats per image row (4096)
constexpr int SEGS = 8;                // vertical segments for occupancy
constexpr int SEG_H = Hn / SEGS;
constexpr float EPS = 1e-8f;

// XOR swizzle of float4 index within a 1024-entry LDS line: makes the
// per-lane serial-chunk access pattern (w = 32*lane + j) hit distinct
// bank groups across lanes instead of a 32-way conflict.
DEV int swz(int w) { return w ^ ((w >> 5) & 15); }

DEV float4 f4zero() { return make_float4(0.f, 0.f, 0.f, 0.f); }
DEV float4 f4add(float4 a, float4 b) { return make_float4(a.x+b.x, a.y+b.y, a.z+b.z, a.w+b.w); }
DEV float4 f4sub(float4 a, float4 b) { return make_float4(a.x-b.x, a.y-b.y, a.z-b.z, a.w-b.w); }
DEV float4 f4mul(float4 a, float4 b) { return make_float4(a.x*b.x, a.y*b.y, a.z*b.z, a.w*b.w); }

// CDNA5 async global->LDS copy (GVS mode: SGPR base + per-lane 32-bit offset).
// LDS destination address is per-lane via the VDST VGPR.
DEV void async_ld_b128(unsigned lds_off, const void* sbase, unsigned voff) {
  asm volatile("global_load_async_to_lds_b128 %0, %1, %2"
               :: "v"(lds_off), "v"(voff), "s"(sbase)
               : "memory");
}
DEV void async_wait0() { asm volatile("s_wait_asynccnt 0" ::: "memory"); }

// Generic pointer to LDS: low 32 bits of the flat address are the LDS offset
// (ISA 10.2: LDS aperture maps addr[31:0] -> LDS offset).
DEV unsigned lds_off_of(const void* p) { return (unsigned)(uintptr_t)p; }

// wave32 exclusive scan of a float4 (per-component), via shuffles.
DEV float4 wave_excl_scan(float4 v, int lane) {
#pragma unroll
  for (int d = 1; d < 32; d <<= 1) {
    float tx = __shfl_up(v.x, d, 32);
    float ty = __shfl_up(v.y, d, 32);
    float tz = __shfl_up(v.z, d, 32);
    float tw = __shfl_up(v.w, d, 32);
    if (lane >= d) { v.x += tx; v.y += ty; v.z += tz; v.w += tw; }
  }
  float ex = __shfl_up(v.x, 1, 32);
  float ey = __shfl_up(v.y, 1, 32);
  float ez = __shfl_up(v.z, 1, 32);
  float ew = __shfl_up(v.w, 1, 32);
  return (lane == 0) ? f4zero() : make_float4(ex, ey, ez, ew);
}

// Clamped window length: matches reference box-of-ones exactly.
DEV float nlen(int i, int L) {
  int hi = min(i + Rr, L - 1);
  int lo = max(i - Rr, 0);
  return (float)(hi - lo + 1);
}

// ---------------------------------------------------------------------------
// Pass A: per image row, horizontal box sums of {I, p, I*p, I*I}.
// One block per row; wave q (of 4) owns quantity q over all 4 channels.
// ws1 layout: [h][q][w][c] (float), q-stride = WC floats.
// ---------------------------------------------------------------------------
__global__ __launch_bounds__(128) void pass_h4(const float* __restrict__ I,
                                               const float* __restrict__ P,
                                               float* __restrict__ ws1) {
  __shared__ float4 sI[Wn];
  __shared__ float4 sP[Wn];
  __shared__ float4 sC[4][Wn];

  const int h    = blockIdx.x;
  const int tid  = threadIdx.x;
  const int lane = tid & 31;
  const int q    = tid >> 5;

  // Stage both input rows into LDS via the async tensor path.
  const float4* gI = (const float4*)I + (size_t)h * Wn;
  const float4* gP = (const float4*)P + (size_t)h * Wn;
  const unsigned baseI = lds_off_of(&sI[0]);
  const unsigned baseP = lds_off_of(&sP[0]);
  for (int k = tid; k < Wn; k += 128) {
    const unsigned dst = (unsigned)swz(k) * 16u;
    const unsigned src = (unsigned)k * 16u;
    async_ld_b128(baseI + dst, (const void*)gI, src);
    async_ld_b128(baseP + dst, (const void*)gP, src);
  }
  async_wait0();
  __syncthreads();

  // Serial per-lane cumsum of 32 contiguous pixels, then wave scan fixup.
  const int w0 = lane * 32;
  float4 run = f4zero();
  for (int j = 0; j < 32; ++j) {
    const int s = swz(w0 + j);
    const float4 a = sI[s];
    const float4 b = sP[s];
    float4 v;
    if (q == 0)      v = a;
    else if (q == 1) v = b;
    else if (q == 2) v = f4mul(a, b);
    else             v = f4mul(a, a);
    run = f4add(run, v);
    sC[q][s] = run;
  }
  const float4 off = wave_excl_scan(run, lane);
  if (lane > 0) {
    for (int j = 0; j < 32; ++j) {
      const int s = swz(w0 + j);
      sC[q][s] = f4add(sC[q][s], off);
    }
  }
  // box[w] = cum[min(w+r, W-1)] - (w-r-1 >= 0 ? cum[w-r-1] : 0)
  float4* o = (float4*)ws1 + ((size_t)h * 4 + q) * Wn;
  for (int j = 0; j < 32; ++j) {
    const int w  = w0 + j;
    const int hi = min(w + Rr, Wn - 1);
    const int lo = w - Rr - 1;
    float4 v = sC[q][swz(hi)];
    if (lo >= 0) v = f4sub(v, sC[q][swz(lo)]);
    o[w] = v;
  }
}

// ---------------------------------------------------------------------------
// Pass B: vertical box (sliding window over h) of the 4 h-boxed quantities,
// fused with the per-pixel statistics -> A, b.  Thread = one (w,c) column,
// segmented in h for occupancy.  ws2 layout: [h][q2][w][c], q2 in {A,b}.
// ---------------------------------------------------------------------------
__global__ __launch_bounds__(256) void pass_v4(const float* __restrict__ ws1,
                                               float* __restrict__ ws2) {
  const int wc = blockIdx.x * 256 + threadIdx.x;  // 0..WC-1
  const int w  = wc >> 2;
  const float nw = nlen(w, Wn);
  const int i0 = blockIdx.y * SEG_H;
  const int i1 = i0 + SEG_H;

  float s0 = 0.f, s1 = 0.f, s2 = 0.f, s3 = 0.f;
  const int hstart = max(0, i0 - Rr);
  for (int hh = hstart; hh < i0 + Rr; ++hh) {       // rows [hstart, i0+r-1]
    const float* p = ws1 + (size_t)hh * 4 * WC + wc;
    s0 += p[0]; s1 += p[WC]; s2 += p[2 * WC]; s3 += p[3 * WC];
  }
  for (int i = i0; i < i1; ++i) {
    if (i + Rr < Hn) {
      const float* p = ws1 + (size_t)(i + Rr) * 4 * WC + wc;
      s0 += p[0]; s1 += p[WC]; s2 += p[2 * WC]; s3 += p[3 * WC];
    }
    const float inv = 1.0f / (nlen(i, Hn) * nw);
    const float mI  = s0 * inv;
    const float mP  = s1 * inv;
    const float cov = s2 * inv - mI * mP;
    const float var = s3 * inv - mI * mI;
    const float A   = cov / (var + EPS);
    const float Bb  = mP - A * mI;
    float* o = ws2 + (size_t)i * 2 * WC + wc;
    o[0]  = A;
    o[WC] = Bb;
    if (i - Rr >= 0) {
      const float* p = ws1 + (size_t)(i - Rr) * 4 * WC + wc;
      s0 -= p[0]; s1 -= p[WC]; s2 -= p[2 * WC]; s3 -= p[3 * WC];
    }
  }
}

// ---------------------------------------------------------------------------
// Pass C: horizontal box of {A, b}.  One block per row, wave q owns A or b.
// ws3 layout: [h][q2][w][c].
// ---------------------------------------------------------------------------
__global__ __launch_bounds__(64) void pass_h2(const float* __restrict__ ws2,
                                              float* __restrict__ ws3) {
  __shared__ float4 sIn[2][Wn];
  __shared__ float4 sC[2][Wn];

  const int h    = blockIdx.x;
  const int tid  = threadIdx.x;
  const int lane = tid & 31;
  const int q    = tid >> 5;

  // The A and b rows are contiguous in ws2: async-stage both (2048 float4s).
  const float4* g = (const float4*)(ws2 + (size_t)h * 2 * WC);
  const unsigned base = lds_off_of(&sIn[0][0]);
  for (int k = tid; k < 2 * Wn; k += 64) {
    const int row = k >> 10;
    const int col = k & (Wn - 1);
    const unsigned dst = (unsigned)(row * Wn + swz(col)) * 16u;
    async_ld_b128(base + dst, (const void*)g, (unsigned)k * 16u);
  }
  async_wait0();
  __syncthreads();

  const int w0 = lane * 32;
  float4 run = f4zero();
  for (int j = 0; j < 32; ++j) {
    const int s = swz(w0 + j);
    run = f4add(run, sIn[q][s]);
    sC[q][s] = run;
  }
  const float4 off = wave_excl_scan(run, lane);
  if (lane > 0) {
    for (int j = 0; j < 32; ++j) {
      const int s = swz(w0 + j);
      sC[q][s] = f4add(sC[q][s], off);
    }
  }
  float4* o = (float4*)ws3 + ((size_t)h * 2 + q) * Wn;
  for (int j = 0; j < 32; ++j) {
    const int w  = w0 + j;
    const int hi = min(w + Rr, Wn - 1);
    const int lo = w - Rr - 1;
    float4 v = sC[q][swz(hi)];
    if (lo >= 0) v = f4sub(v, sC[q][swz(lo)]);
    o[w] = v;
  }
}

// ---------------------------------------------------------------------------
// Pass D: vertical box of {A, b} fused with q = mean_A * I + mean_b.
// Non-temporal stores: q is never re-read, keep L2 for the intermediates.
// ---------------------------------------------------------------------------
__global__ __launch_bounds__(256) void pass_vfinal(const float* __restrict__ ws3,
                                                   const float* __restrict__ I,
                                                   float* __restrict__ out) {
  const int wc = blockIdx.x * 256 + threadIdx.x;
  const int w  = wc >> 2;
  const float nw = nlen(w, Wn);
  const int i0 = blockIdx.y * SEG_H;
  const int i1 = i0 + SEG_H;

  float sA = 0.f, sB = 0.f;
  const int hstart = max(0, i0 - Rr);
  for (int hh = hstart; hh < i0 + Rr; ++hh) {
    const float* p = ws3 + (size_t)hh * 2 * WC + wc;
    sA += p[0]; sB += p[WC];
  }
  for (int i = i0; i < i1; ++i) {
    if (i + Rr < Hn) {
      const float* p = ws3 + (size_t)(i + Rr) * 2 * WC + wc;
      sA += p[0]; sB += p[WC];
    }
    const float inv = 1.0f / (nlen(i, Hn) * nw);
    const float val = I[(size_t)i * WC + wc];
    const float qv  = (sA * inv) * val + (sB * inv);
    __builtin_nontemporal_store(qv, out + (size_t)i * WC + wc);
    if (i - Rr >= 0) {
      const float* p = ws3 + (size_t)(i - Rr) * 2 * WC + wc;
      sA -= p[0]; sB -= p[WC];
    }
  }
}

} // namespace gf

extern "C" void kernel_launch(void* const* d_in, const int* in_sizes, int n_in,
                              void* d_out, int out_size, void* d_ws, size_t ws_size,
                              hipStream_t stream) {
  using namespace gf;
  (void)in_sizes; (void)n_in; (void)out_size; (void)ws_size;

  const float* I = (const float*)d_in[0];
  const float* P = (const float*)d_in[1];
  // d_in[2] is r (== 40), baked in as Rr.
  float* out = (float*)d_out;

  // Per-batch workspace (processed sequentially so it stays L2-resident):
  //   ws1: H * 4 * W * C floats (16,777,216 floats = 64 MiB), reused as ws3
  //   ws2: H * 2 * W * C floats ( 8,388,608 floats = 32 MiB)
  float* ws1 = (float*)d_ws;
  float* ws2 = ws1 + (size_t)Hn * 4 * WC;

  for (int b = 0; b < Bn; ++b) {
    const float* Ib = I   + (size_t)b * Hn * WC;
    const float* Pb = P   + (size_t)b * Hn * WC;
    float*       Ob = out + (size_t)b * Hn * WC;

    gf::pass_h4    <<<dim3(Hn),          dim3(128), 0, stream>>>(Ib, Pb, ws1);
    gf::pass_v4    <<<dim3(WC/256, SEGS), dim3(256), 0, stream>>>(ws1, ws2);
    gf::pass_h2    <<<dim3(Hn),          dim3(64),  0, stream>>>(ws2, ws1);
    gf::pass_vfinal<<<dim3(WC/256, SEGS), dim3(256), 0, stream>>>(ws1, Ib, Ob);
  }
}